// GATRegressor_39341900431369
// MI455X (gfx1250) — compile-verified
//
#include <hip/hip_runtime.h>
#include <hip/hip_bf16.h>

// ---------------------------------------------------------------------------
// GAT regressor pipeline for MI455X (gfx1250, wave32).
// GEMMs use V_WMMA_F32_16X16X4_F32 (full fp32 precision on the matrix pipe),
// with the shared A stripe staged through LDS (once per block instead of once
// per wave). Edge softmax-aggregation is the memory-bound part; atomics.
// ---------------------------------------------------------------------------

typedef __attribute__((ext_vector_type(2))) float v2f;
typedef __attribute__((ext_vector_type(8))) float v8f;

#define N_HEADS 2
#define CH      128   // HEADS*HID
#define HID_C   64
#define APAD    4     // LDS row padding: bank = 4*m + k (mod 64), conflict-free

// ---------------- fp32 WMMA GEMM: H[n,128] = A[n,128] @ W[128,128] ----------
// One 16x16 output tile per wave; 8 waves per block cover the 8 column tiles
// of one 16-row stripe. nRows must be a multiple of 16 (50000 = 3125*16).
__global__ __launch_bounds__(256) void gemm128_wmma(
    const float* __restrict__ A, const float* __restrict__ W,
    float* __restrict__ H) {
  __shared__ float As[16][CH + APAD];
  const int tid  = threadIdx.x;
  const int lane = tid & 31;
  const int wave = tid >> 5;                  // 0..7 => column tile
  const int row0 = blockIdx.x * 16;

  // cooperative stage: 16 rows x 128 f32 = 8KB, 8 floats per thread
  {
    const int r  = tid >> 4;                  // 0..15
    const int c0 = (tid & 15) * 8;            // 0..120
    const float4* srcv = (const float4*)(A + (size_t)(row0 + r) * CH + c0);
    float4 v0 = srcv[0], v1 = srcv[1];
    *(float4*)&As[r][c0]     = v0;
    *(float4*)&As[r][c0 + 4] = v1;
  }
  __syncthreads();

  const int col0 = wave * 16;
  const int mn   = lane & 15;                 // M index for A, N index for B/C
  const int kk   = (lane >> 4) << 1;          // 0 or 2: K sub-offset per lane half

  v8f c = {};
  for (int k0 = 0; k0 < CH; k0 += 4) {
    v2f a, b;
    // A 16x4 fragment from LDS: lanes 0-15 K={k0,k0+1}, lanes 16-31 K={k0+2,k0+3}
    a.x = As[mn][k0 + kk];
    a.y = As[mn][k0 + kk + 1];
    // B 4x16 fragment (row K striped across lanes)
    b.x = W[(size_t)(k0 + kk) * CH + col0 + mn];
    b.y = W[(size_t)(k0 + kk + 1) * CH + col0 + mn];
    c = __builtin_amdgcn_wmma_f32_16x16x4_f32(
        /*neg_a=*/false, a, /*neg_b=*/false, b,
        /*c_mod=*/(short)0, c, /*reuse_a=*/false, /*reuse_b=*/false);
  }
  // C/D layout: VGPR r -> lanes 0-15: (M=r, N=lane), lanes 16-31: (M=8+r)
  const int mb = (lane >> 4) * 8;
  float* orow = H + (size_t)(row0 + mb) * CH + col0 + mn;
#pragma unroll
  for (int r = 0; r < 8; ++r) orow[(size_t)r * CH] = c[r];
}

// ------------- per-node attention logits: alpha[n,4] = {as0,as1,ad0,ad1} ----
__global__ void alpha_kernel(const float* __restrict__ H,
                             const float* __restrict__ asrc,
                             const float* __restrict__ adst,
                             float* __restrict__ alpha, int n) {
  int t = blockIdx.x * blockDim.x + threadIdx.x;
  if (t >= n * N_HEADS) return;
  int node = t >> 1, h = t & 1;
  const float* hv = H + (size_t)node * CH + h * HID_C;
  const float* as = asrc + h * HID_C;
  const float* ad = adst + h * HID_C;
  float s = 0.f, d = 0.f;
#pragma unroll 4
  for (int c = 0; c < HID_C; ++c) {
    float v = hv[c];
    s += v * as[c];
    d += v * ad[c];
  }
  alpha[(size_t)node * 4 + h] = s;
  alpha[(size_t)node * 4 + 2 + h] = d;
}

// order-preserving float <-> uint encoding for atomicMax-based segment max
__device__ __forceinline__ unsigned fenc(float f) {
  unsigned u = __float_as_uint(f);
  return (u & 0x80000000u) ? ~u : (u | 0x80000000u);
}
__device__ __forceinline__ float fdec(unsigned u) {
  return (u & 0x80000000u) ? __uint_as_float(u & 0x7fffffffu)
                           : __uint_as_float(~u);
}

// pass 1: logits = leaky_relu(alpha_s[src]+alpha_d[dst]); segment max by dst
__global__ void edge_logits_max(const int* __restrict__ src,
                                const int* __restrict__ dst,
                                const float* __restrict__ alpha,
                                float* __restrict__ logits,
                                unsigned* __restrict__ menc, int nE, int nN) {
  int e = blockIdx.x * blockDim.x + threadIdx.x;
  if (e >= nE + nN) return;
  int s = (e < nE) ? src[e] : (e - nE);
  int d = (e < nE) ? dst[e] : (e - nE);
#pragma unroll
  for (int h = 0; h < N_HEADS; ++h) {
    float l = alpha[(size_t)s * 4 + h] + alpha[(size_t)d * 4 + 2 + h];
    l = (l > 0.f) ? l : 0.2f * l;           // leaky relu
    logits[(size_t)e * 2 + h] = l;
    atomicMax(&menc[(size_t)d * 2 + h], fenc(l));
  }
}

// pass 2: e = exp(logit - max[dst]); segment sum by dst; exp stored in place
__global__ void edge_exp_sum(const int* __restrict__ src,
                             const int* __restrict__ dst,
                             float* __restrict__ logits,
                             const unsigned* __restrict__ menc,
                             float* __restrict__ ssum, int nE, int nN) {
  int e = blockIdx.x * blockDim.x + threadIdx.x;
  if (e >= nE + nN) return;
  int d = (e < nE) ? dst[e] : (e - nE);
#pragma unroll
  for (int h = 0; h < N_HEADS; ++h) {
    float ex = __expf(logits[(size_t)e * 2 + h] - fdec(menc[(size_t)d * 2 + h]));
    logits[(size_t)e * 2 + h] = ex;
    atomicAdd(&ssum[(size_t)d * 2 + h], ex);
  }
}

// pass 3: OUT[dst] += h[src] * (e / s[dst]); thread = (edge, 4-channel group)
__global__ void edge_aggregate(const int* __restrict__ src,
                               const int* __restrict__ dst,
                               const float* __restrict__ H,
                               const float* __restrict__ ebuf,
                               const float* __restrict__ ssum,
                               float* __restrict__ OUT, int nE, int nN) {
  long long t = blockIdx.x * (long long)blockDim.x + threadIdx.x;
  long long tot = (long long)(nE + nN) * 32;
  if (t >= tot) return;
  int e = (int)(t >> 5);
  int g = (int)(t & 31);                     // channels [g*4, g*4+4)
  int s = (e < nE) ? src[e] : (e - nE);
  int d = (e < nE) ? dst[e] : (e - nE);
  int h = g >> 4;                            // channels 0-63 head0, 64-127 head1
  float w = ebuf[(size_t)e * 2 + h] / ssum[(size_t)d * 2 + h];
  const float4 v = ((const float4*)(H + (size_t)s * CH))[g];
  float* o = OUT + (size_t)d * CH + g * 4;
  atomicAdd(o + 0, v.x * w);
  atomicAdd(o + 1, v.y * w);
  atomicAdd(o + 2, v.z * w);
  atomicAdd(o + 3, v.w * w);
}

// bias + optional relu
__global__ void bias_act(const float* __restrict__ IN,
                         const float* __restrict__ b, float* __restrict__ OUT,
                         long long total, int doRelu) {
  long long t = blockIdx.x * (long long)blockDim.x + threadIdx.x;
  if (t >= total) return;
  float v = IN[t] + b[(int)(t & (CH - 1))];
  OUT[t] = doRelu ? fmaxf(v, 0.f) : v;
}

// mean-pool accumulation over graphs
__global__ void pool_kernel(const float* __restrict__ H,
                            const int* __restrict__ batch,
                            float* __restrict__ pooled,
                            float* __restrict__ cnt, int n) {
  long long t = blockIdx.x * (long long)blockDim.x + threadIdx.x;
  if (t >= (long long)n * CH) return;
  int node = (int)(t >> 7), c = (int)(t & (CH - 1));
  int g = batch[node];
  atomicAdd(&pooled[(size_t)g * CH + c], H[t]);
  if (c == 0) atomicAdd(&cnt[g], 1.0f);
}

// pooled mean -> lin1 (128->64) -> relu -> lin2 (64->1); one thread per graph
__global__ void head_mlp(const float* __restrict__ pooled,
                         const float* __restrict__ cnt,
                         const float* __restrict__ w1,
                         const float* __restrict__ b1,
                         const float* __restrict__ w2,
                         const float* __restrict__ b2,
                         float* __restrict__ out, int nG) {
  int g = blockIdx.x * blockDim.x + threadIdx.x;
  if (g >= nG) return;
  float inv = 1.0f / fmaxf(cnt[g], 1.0f);
  float acc = 0.f;
  for (int j = 0; j < HID_C; ++j) {
    float hj = b1[j];
    for (int k = 0; k < CH; ++k)
      hj += pooled[(size_t)g * CH + k] * inv * w1[(size_t)k * HID_C + j];
    hj = fmaxf(hj, 0.f);
    acc += hj * w2[j];
  }
  out[g] = acc + b2[0];
}

extern "C" void kernel_launch(void* const* d_in, const int* in_sizes, int n_in,
                              void* d_out, int out_size, void* d_ws,
                              size_t ws_size, hipStream_t stream) {
  const float* x       = (const float*)d_in[0];
  const int*   eidx    = (const int*)d_in[1];
  const int*   batch   = (const int*)d_in[3];
  const float* Wk[4]   = {(const float*)d_in[4],  (const float*)d_in[8],
                          (const float*)d_in[12], (const float*)d_in[16]};
  const float* asrc[4] = {(const float*)d_in[5],  (const float*)d_in[9],
                          (const float*)d_in[13], (const float*)d_in[17]};
  const float* adst[4] = {(const float*)d_in[6],  (const float*)d_in[10],
                          (const float*)d_in[14], (const float*)d_in[18]};
  const float* bk[4]   = {(const float*)d_in[7],  (const float*)d_in[11],
                          (const float*)d_in[15], (const float*)d_in[19]};
  const float* l1w = (const float*)d_in[20];
  const float* l1b = (const float*)d_in[21];
  const float* l2w = (const float*)d_in[22];
  const float* l2b = (const float*)d_in[23];

  const int nN = in_sizes[3];            // 50000 nodes
  const int nE = in_sizes[1] / 2;        // 800000 edges
  const int nA = nE + nN;                // + self loops
  const int nG = out_size;               // 64 graphs
  const int* srcE = eidx;
  const int* dstE = eidx + nE;

  // ---- carve workspace (all offsets 16B aligned) ----
  const size_t NH = (size_t)nN * CH;
  char* p = (char*)d_ws;
  float*    hbuf   = (float*)p;    p += NH * 4;                 // gemm output h
  float*    aggbuf = (float*)p;    p += NH * 4;                 // aggregation out
  float*    actbuf = (float*)p;    p += NH * 4;                 // post-activation
  float*    alphab = (float*)p;    p += (size_t)nN * 4 * 4;     // [N,4]
  unsigned* menc   = (unsigned*)p; p += (size_t)nN * 2 * 4;     // encoded max
  float*    sbuf   = (float*)p;    p += (size_t)nN * 2 * 4;     // softmax denom
  float*    logits = (float*)p;    p += (size_t)nA * 2 * 4;     // logits -> exp
  float*    pooled = (float*)p;    p += (size_t)nG * CH * 4;
  float*    cnt    = (float*)p;    p += (size_t)nG * 4;
  (void)ws_size; (void)n_in;

  const int TB = 256;
  const int gemmBlocks = nN / 16;                         // 3125
  const int edgeBlocks = (nA + TB - 1) / TB;
  const long long aggT = (long long)nA * 32;
  const int aggBlocks  = (int)((aggT + TB - 1) / TB);
  const long long nodeT = (long long)nN * CH;
  const int nodeBlocks  = (int)((nodeT + TB - 1) / TB);

  const float* layerIn = x;
  for (int k = 0; k < 4; ++k) {
    // h = layerIn @ W_k  (fp32 WMMA, A stripe staged through LDS)
    gemm128_wmma<<<gemmBlocks, TB, 0, stream>>>(layerIn, Wk[k], hbuf);
    // per-node attention coefficients
    alpha_kernel<<<(nN * 2 + TB - 1) / TB, TB, 0, stream>>>(hbuf, asrc[k],
                                                            adst[k], alphab, nN);
    // segment softmax over incoming edges
    hipMemsetAsync(menc, 0, (size_t)nN * 2 * 4, stream);  // enc-space -inf
    hipMemsetAsync(sbuf, 0, (size_t)nN * 2 * 4, stream);
    edge_logits_max<<<edgeBlocks, TB, 0, stream>>>(srcE, dstE, alphab, logits,
                                                   menc, nE, nN);
    edge_exp_sum<<<edgeBlocks, TB, 0, stream>>>(srcE, dstE, logits, menc, sbuf,
                                                nE, nN);
    // weighted scatter aggregation
    hipMemsetAsync(aggbuf, 0, NH * 4, stream);
    edge_aggregate<<<aggBlocks, TB, 0, stream>>>(srcE, dstE, hbuf, logits,
                                                 sbuf, aggbuf, nE, nN);
    // bias (+ relu on layers 1-3)
    bias_act<<<nodeBlocks, TB, 0, stream>>>(aggbuf, bk[k], actbuf, nodeT,
                                            k < 3 ? 1 : 0);
    layerIn = actbuf;
  }

  // global mean pool + MLP head
  hipMemsetAsync(pooled, 0, (size_t)nG * CH * 4, stream);
  hipMemsetAsync(cnt, 0, (size_t)nG * 4, stream);
  pool_kernel<<<nodeBlocks, TB, 0, stream>>>(actbuf, batch, pooled, cnt, nN);
  head_mlp<<<(nG + 63) / 64, 64, 0, stream>>>(pooled, cnt, l1w, l1b, l2w, l2b,
                                              (float*)d_out, nG);
}